// CorrBlock_cascade_37744172597294
// MI455X (gfx1250) — compile-verified
//
#include <hip/hip_runtime.h>
#include <hip/hip_bf16.h>

// ---------------------------------------------------------------------------
// CorrBlock cascade for MI455X (gfx1250, wave32, WMMA, async-to-LDS).
// Pipeline (all fp32, memory-bound => keep full precision, use f32 WMMA):
//   K0: zero stats buffer
//   K1: GEMM1  y = x @ w21^T               (v_wmma_f32_16x16x4_f32, K=64)
//   K2: per-channel sum/sumsq of y  (16 ch), float4 sweeps
//   K3: cost volume: a,b = relu(bn(y)), temporal shift, 49-offset corr
//       (halo tile staged with GLOBAL_LOAD_ASYNC_TO_LDS_B32, transformed
//        in place in LDS; LDS float atomics fold corr BN stats)
//   K4: GEMM2  out_pre = relu(bn(corr)) @ w22^T   (WMMA, K=49 padded to 52)
//   K2: per-channel sum/sumsq of out_pre (64 ch)
//   K5: out = relu(bn(out_pre) + x), float4 sweep
// ---------------------------------------------------------------------------

#define NT    64
#define CH    64
#define CMID  16
#define HH    56
#define WW    56
#define HW    3136          // 56*56  (divisible by 16)
#define KK    49
#define EPSB  1e-5f
#define TILES_PER_IMG 196   // HW/16
#define NTILES 12544        // NT * TILES_PER_IMG
#define NPIX  200704        // NT*HW  (per-channel BN count for all three BNs)

#define SOFF_Y 0            // stats: 16 ch * {sum,sumsq}
#define SOFF_K 32           // stats: 49 ch
#define SOFF_O 130          // stats: 64 ch
#define STATS_LEN 258

typedef __attribute__((ext_vector_type(2))) float v2f;
typedef __attribute__((ext_vector_type(4))) float v4f;
typedef __attribute__((ext_vector_type(8))) float v8f;

__device__ __forceinline__ void bn_affine(const float* __restrict__ stats, int soff,
                                          int c, float g, float b,
                                          float& sc, float& bi) {
    const float inv_cnt = 1.0f / (float)NPIX;
    float m  = stats[soff + 2 * c] * inv_cnt;
    float vq = stats[soff + 2 * c + 1] * inv_cnt;
    float var = vq - m * m;
    float rstd = rsqrtf(var + EPSB);
    sc = rstd * g;
    bi = b - m * sc;
}

// --------------------------- K0: zero stats --------------------------------
__global__ void init_stats_kernel(float* __restrict__ stats) {
    int i = threadIdx.x;
    if (i < STATS_LEN) stats[i] = 0.0f;
}

// --------------------------- K1: GEMM1 via WMMA ----------------------------
// y[n,o,p] = sum_c x[n,c,p] * w21[o,c];   tile = 16 pixels, N = 16 outputs,
// K = 64 in 16 steps of V_WMMA_F32_16X16X4_F32.
__global__ void __launch_bounds__(256)
gemm1_wmma_kernel(const float* __restrict__ x, const float* __restrict__ w21,
                  float* __restrict__ y_ws) {
    const int wave = threadIdx.x >> 5;
    const int lane = threadIdx.x & 31;
    const int tile = blockIdx.x * 8 + wave;          // 0..NTILES-1
    if (tile >= NTILES) return;

    const int n     = tile / TILES_PER_IMG;
    const int pbase = (tile % TILES_PER_IMG) * 16;
    const int mrow  = lane & 15;                     // A: pixel row, B: out ch
    const int hi    = lane >> 4;                     // K sub-pair selector

    const float* __restrict__ xn = x + (size_t)n * CH * HW;

    v8f acc = {};
#pragma unroll
    for (int kb = 0; kb < 16; ++kb) {
        const int c0 = kb * 4 + hi * 2;
        v2f a, b;
        // A (16x4, MxK): lane holds pixel mrow, channels c0,c0+1
        a.x = xn[(size_t)(c0 + 0) * HW + pbase + mrow];
        a.y = xn[(size_t)(c0 + 1) * HW + pbase + mrow];
        // B (4x16, KxN): lane holds output channel mrow, rows c0,c0+1
        b.x = w21[mrow * CH + c0 + 0];
        b.y = w21[mrow * CH + c0 + 1];
        acc = __builtin_amdgcn_wmma_f32_16x16x4_f32(
                false, a, false, b, (short)0, acc, false, false);
    }

    // D: lane (hi,mrow) holds output channel o=mrow, pixels pbase+hi*8+v
    // pbase % 16 == 0, HW % 16 == 0 -> 32B aligned -> two b128 stores
    const int o = mrow;
    float* __restrict__ yrow = y_ws + ((size_t)n * CMID + o) * HW + pbase + hi * 8;
    v4f st0 = { acc[0], acc[1], acc[2], acc[3] };
    v4f st1 = { acc[4], acc[5], acc[6], acc[7] };
    *(v4f*)(yrow)     = st0;
    *(v4f*)(yrow + 4) = st1;
}

// ------------------- K2: per-channel sum / sumsq reduction -----------------
// data layout [NT, Cc, HW]; grid = (slabs, Cc); float4 (b128) sweeps.
__global__ void __launch_bounds__(256)
chan_stats_kernel(const float* __restrict__ data, float* __restrict__ stats,
                  int Cc, int soff) {
    const int c = blockIdx.y;
    const int HW4 = HW / 4;           // 784
    const int total4 = NPIX / 4;      // 50176
    float s = 0.0f, q = 0.0f;
    const int stride = gridDim.x * blockDim.x;
    for (int i = blockIdx.x * blockDim.x + threadIdx.x; i < total4; i += stride) {
        const int n = i / HW4, p4 = i - n * HW4;
        const v4f v = *(const v4f*)(data + ((size_t)n * Cc + c) * HW + p4 * 4);
        s += v.x + v.y + v.z + v.w;
        q += v.x * v.x + v.y * v.y + v.z * v.z + v.w * v.w;
    }
    __shared__ float rs[256], rq[256];
    rs[threadIdx.x] = s;
    rq[threadIdx.x] = q;
    __syncthreads();
    for (int off = 128; off > 0; off >>= 1) {
        if (threadIdx.x < off) {
            rs[threadIdx.x] += rs[threadIdx.x + off];
            rq[threadIdx.x] += rq[threadIdx.x + off];
        }
        __syncthreads();
    }
    if (threadIdx.x == 0) {
        atomicAdd(&stats[soff + 2 * c], rs[0]);
        atomicAdd(&stats[soff + 2 * c + 1], rq[0]);
    }
}

// --------------------------- K3: cost volume -------------------------------
// a = relu(bn(y[n]));  b = relu(bn(y[nsrc])) with temporal shift; 49 offsets.
// Halo tile 22x22x16 staged into LDS with GLOBAL_LOAD_ASYNC_TO_LDS_B32
// (ASYNCcnt path, bypasses VGPRs), then bn+relu applied in place; padding
// entries are written 0 by regular ds stores (disjoint address sets).
__global__ void __launch_bounds__(256)
corr_kernel(const float* __restrict__ y_ws, const float* __restrict__ g21,
            const float* __restrict__ b21, float* __restrict__ corr_ws,
            float* __restrict__ stats) {
    const int n  = blockIdx.y;
    const int tx = blockIdx.x & 3;
    const int ty = blockIdx.x >> 2;
    const int tid = threadIdx.x;

    __shared__ float y_sc[CMID], y_bi[CMID];
    __shared__ float lds_b[CMID * 22 * 22];   // 7744 floats
    __shared__ float ksum[KK], ksq[KK];

    if (tid < CMID) bn_affine(stats, SOFF_Y, tid, g21[tid], b21[tid],
                              y_sc[tid], y_bi[tid]);
    if (tid < KK) { ksum[tid] = 0.0f; ksq[tid] = 0.0f; }

    // temporal shift: n = b*8 + t, source time = min(t+1, 7)
    const int t = n & 7;
    const int nsrc = (t == 7) ? n : (n + 1);
    const float* __restrict__ yb = y_ws + (size_t)nsrc * CMID * HW;

    // Stage raw y tile: async-to-LDS for in-bounds, zero ds_store for padding.
    for (int idx = tid; idx < CMID * 22 * 22; idx += 256) {
        const int c  = idx / 484;
        const int r  = idx - c * 484;
        const int yy = r / 22, xx = r - (r / 22) * 22;
        const int gy = ty * 16 - 3 + yy;
        const int gx = tx * 16 - 3 + xx;
        if (gy >= 0 && gy < HH && gx >= 0 && gx < WW) {
            const float* gp = yb + (size_t)c * HW + gy * WW + gx;
            const unsigned lds_addr = (unsigned)(uintptr_t)(&lds_b[idx]);
            asm volatile("global_load_async_to_lds_b32 %0, %1, off"
                         :: "v"(lds_addr), "v"(gp)
                         : "memory");
        } else {
            lds_b[idx] = 0.0f;
        }
    }
    asm volatile("s_wait_asynccnt 0x0" ::: "memory");
    __syncthreads();

    // In-place bn+relu on in-bounds entries only (padding must stay 0).
    for (int idx = tid; idx < CMID * 22 * 22; idx += 256) {
        const int c  = idx / 484;
        const int r  = idx - c * 484;
        const int yy = r / 22, xx = r - (r / 22) * 22;
        const int gy = ty * 16 - 3 + yy;
        const int gx = tx * 16 - 3 + xx;
        if (gy >= 0 && gy < HH && gx >= 0 && gx < WW) {
            lds_b[idx] = fmaxf(lds_b[idx] * y_sc[c] + y_bi[c], 0.0f);
        }
    }
    __syncthreads();

    const int ly = tid >> 4, lx = tid & 15;
    const int i = ty * 16 + ly, j = tx * 16 + lx;
    const bool valid = (i < HH) && (j < WW);

    if (valid) {
        const float* __restrict__ ya = y_ws + (size_t)n * CMID * HW;
        float a[CMID];
#pragma unroll
        for (int c = 0; c < CMID; ++c) {
            float v = ya[(size_t)c * HW + i * WW + j] * y_sc[c] + y_bi[c];
            a[c] = fmaxf(v, 0.0f);
        }
#pragma unroll
        for (int dy = 0; dy < 7; ++dy) {
#pragma unroll
            for (int dx = 0; dx < 7; ++dx) {
                float s = 0.0f;
#pragma unroll
                for (int c = 0; c < CMID; ++c)
                    s += a[c] * lds_b[c * 484 + (ly + dy) * 22 + (lx + dx)];
                s *= (1.0f / (float)CMID);
                const int k = dy * 7 + dx;
                corr_ws[((size_t)n * KK + k) * HW + i * WW + j] = s;
                atomicAdd(&ksum[k], s);        // ds_add_f32
                atomicAdd(&ksq[k], s * s);
            }
        }
    }
    __syncthreads();
    if (tid < KK) {
        atomicAdd(&stats[SOFF_K + 2 * tid], ksum[tid]);
        atomicAdd(&stats[SOFF_K + 2 * tid + 1], ksq[tid]);
    }
}

// --------------------------- K4: GEMM2 via WMMA ----------------------------
// out_pre[n,o,p] = sum_k relu(bn(corr[n,k,p])) * w22[o,k];  K=49 -> 13 steps,
// 4 output groups of 16 sharing one A tile.
__global__ void __launch_bounds__(256)
gemm2_wmma_kernel(const float* __restrict__ corr_ws, const float* __restrict__ w22,
                  const float* __restrict__ g22, const float* __restrict__ b22,
                  const float* __restrict__ stats, float* __restrict__ outp_ws) {
    __shared__ float csc[52], cbi[52];
    if (threadIdx.x < KK) {
        bn_affine(stats, SOFF_K, threadIdx.x, g22[threadIdx.x], b22[threadIdx.x],
                  csc[threadIdx.x], cbi[threadIdx.x]);
    } else if (threadIdx.x < 52) {
        csc[threadIdx.x] = 0.0f;   // K padding -> relu(0*c + 0) = 0
        cbi[threadIdx.x] = 0.0f;
    }
    __syncthreads();

    const int wave = threadIdx.x >> 5;
    const int lane = threadIdx.x & 31;
    const int tile = blockIdx.x * 8 + wave;
    if (tile >= NTILES) return;

    const int n     = tile / TILES_PER_IMG;
    const int pbase = (tile % TILES_PER_IMG) * 16;
    const int mrow  = lane & 15;
    const int hi    = lane >> 4;

    const float* __restrict__ cn = corr_ws + (size_t)n * KK * HW;

    v8f acc0 = {}, acc1 = {}, acc2 = {}, acc3 = {};
#pragma unroll
    for (int kb = 0; kb < 13; ++kb) {
        const int k0 = kb * 4 + hi * 2;
        v2f a;
        float r0 = (k0 < KK)     ? cn[(size_t)k0 * HW + pbase + mrow]       : 0.0f;
        float r1 = (k0 + 1 < KK) ? cn[(size_t)(k0 + 1) * HW + pbase + mrow] : 0.0f;
        a.x = fmaxf(r0 * csc[k0] + cbi[k0], 0.0f);
        a.y = fmaxf(r1 * csc[k0 + 1] + cbi[k0 + 1], 0.0f);

        v2f b;
        // og = 0..3: output channels og*16 + mrow
        b.x = (k0 < KK)     ? w22[(0 * 16 + mrow) * KK + k0]     : 0.0f;
        b.y = (k0 + 1 < KK) ? w22[(0 * 16 + mrow) * KK + k0 + 1] : 0.0f;
        acc0 = __builtin_amdgcn_wmma_f32_16x16x4_f32(false, a, false, b, (short)0, acc0, false, false);
        b.x = (k0 < KK)     ? w22[(1 * 16 + mrow) * KK + k0]     : 0.0f;
        b.y = (k0 + 1 < KK) ? w22[(1 * 16 + mrow) * KK + k0 + 1] : 0.0f;
        acc1 = __builtin_amdgcn_wmma_f32_16x16x4_f32(false, a, false, b, (short)0, acc1, false, false);
        b.x = (k0 < KK)     ? w22[(2 * 16 + mrow) * KK + k0]     : 0.0f;
        b.y = (k0 + 1 < KK) ? w22[(2 * 16 + mrow) * KK + k0 + 1] : 0.0f;
        acc2 = __builtin_amdgcn_wmma_f32_16x16x4_f32(false, a, false, b, (short)0, acc2, false, false);
        b.x = (k0 < KK)     ? w22[(3 * 16 + mrow) * KK + k0]     : 0.0f;
        b.y = (k0 + 1 < KK) ? w22[(3 * 16 + mrow) * KK + k0 + 1] : 0.0f;
        acc3 = __builtin_amdgcn_wmma_f32_16x16x4_f32(false, a, false, b, (short)0, acc3, false, false);
    }

#pragma unroll
    for (int og = 0; og < 4; ++og) {
        const v8f acc = (og == 0) ? acc0 : (og == 1) ? acc1 : (og == 2) ? acc2 : acc3;
        const int o = og * 16 + mrow;
        float* __restrict__ orow = outp_ws + ((size_t)n * CH + o) * HW + pbase + hi * 8;
        v4f st0 = { acc[0], acc[1], acc[2], acc[3] };
        v4f st1 = { acc[4], acc[5], acc[6], acc[7] };
        *(v4f*)(orow)     = st0;
        *(v4f*)(orow + 4) = st1;
    }
}

// --------------------------- K5: final BN + residual + relu ----------------
// float4 sweep; HW % 4 == 0 so the channel is uniform within a float4.
__global__ void __launch_bounds__(256)
final_kernel(const float* __restrict__ outp_ws, const float* __restrict__ x,
             const float* __restrict__ g23, const float* __restrict__ b23,
             const float* __restrict__ stats, float* __restrict__ out) {
    const size_t idx4 = (size_t)blockIdx.x * 256 + threadIdx.x;
    const size_t total4 = (size_t)NT * CH * HW / 4;
    if (idx4 >= total4) return;
    const int o = (int)((idx4 * 4 / HW) % CH);
    float sc, bi;
    bn_affine(stats, SOFF_O, o, g23[o], b23[o], sc, bi);
    const v4f ov = *(const v4f*)(outp_ws + idx4 * 4);
    const v4f xv = *(const v4f*)(x + idx4 * 4);
    v4f r;
    r.x = fmaxf(ov.x * sc + bi + xv.x, 0.0f);
    r.y = fmaxf(ov.y * sc + bi + xv.y, 0.0f);
    r.z = fmaxf(ov.z * sc + bi + xv.z, 0.0f);
    r.w = fmaxf(ov.w * sc + bi + xv.w, 0.0f);
    *(v4f*)(out + idx4 * 4) = r;
}

// ---------------------------------------------------------------------------
extern "C" void kernel_launch(void* const* d_in, const int* in_sizes, int n_in,
                              void* d_out, int out_size, void* d_ws, size_t ws_size,
                              hipStream_t stream) {
    const float* x   = (const float*)d_in[0];
    const float* w21 = (const float*)d_in[1];
    const float* w22 = (const float*)d_in[2];
    const float* g21 = (const float*)d_in[3];
    const float* b21 = (const float*)d_in[4];
    const float* g22 = (const float*)d_in[5];
    const float* b22 = (const float*)d_in[6];
    const float* g23 = (const float*)d_in[7];
    const float* b23 = (const float*)d_in[8];
    float* out = (float*)d_out;

    float* y_ws    = (float*)d_ws;                         // 64*16*3136
    float* corr_ws = y_ws + (size_t)NT * CMID * HW;        // 64*49*3136
    float* outp_ws = corr_ws + (size_t)NT * KK * HW;       // 64*64*3136
    float* stats   = outp_ws + (size_t)NT * CH * HW;       // 258 floats

    init_stats_kernel<<<1, 256, 0, stream>>>(stats);

    gemm1_wmma_kernel<<<NTILES / 8, 256, 0, stream>>>(x, w21, y_ws);

    chan_stats_kernel<<<dim3(64, CMID), 256, 0, stream>>>(y_ws, stats, CMID, SOFF_Y);

    corr_kernel<<<dim3(16, NT), 256, 0, stream>>>(y_ws, g21, b21, corr_ws, stats);

    gemm2_wmma_kernel<<<NTILES / 8, 256, 0, stream>>>(corr_ws, w22, g22, b22,
                                                      stats, outp_ws);

    chan_stats_kernel<<<dim3(64, CH), 256, 0, stream>>>(outp_ws, stats, CH, SOFF_O);

    const int total4 = NT * CH * HW / 4;
    final_kernel<<<(total4 + 255) / 256, 256, 0, stream>>>(outp_ws, x, g23, b23,
                                                           stats, out);
}